// AttnDecoderRNN_2130303779125
// MI455X (gfx1250) — compile-verified
//
#include <hip/hip_runtime.h>
#include <math.h>

#define H 1024
#define V 50257
#define L 100

typedef float v2f __attribute__((ext_vector_type(2)));
typedef float v8f __attribute__((ext_vector_type(8)));
typedef int   v4i __attribute__((vector_size(16)));   // matches builtin param type

#define AS1 __attribute__((address_space(1)))
#define AS3 __attribute__((address_space(3)))

#if __has_builtin(__builtin_amdgcn_global_load_async_to_lds_b128)
#define GEMV_ASYNC 1
#else
#define GEMV_ASYNC 0
#endif

__device__ __forceinline__ AS1 v4i* to_as1(const void* p) {
    return (AS1 v4i*)(uintptr_t)p;            // global: generic addr == AS1 addr
}
__device__ __forceinline__ AS3 v4i* to_as3(const void* p) {
    // generic shared addr: low 32 bits are the wave-relative LDS byte offset
    return (AS3 v4i*)(uint32_t)(uintptr_t)p;
}

// ---------------------------------------------------------------------------
// WMMA-based GEMV: y[m] = act( sum_k W[m,k] * x[k] + bias[m] ),  W row-major.
// One wave (32 threads) per block; each block computes a 16-row tile.
// K must be a multiple of 128. M arbitrary (rows clamped on load, guarded on
// store). Per 128-col chunk: one coalesced b128 load per row per wave into an
// LDS tile (row stride 132 floats = 528B: 16B-aligned, bank-conflict-free),
// then 32 V_WMMA_F32_16X16X4_F32 steps. The x-vector occupies column 0 of B;
// D column 0 (lanes 0 and 16) holds the 16 dot products.
// Double-buffered: chunk k+1 is fetched (async-to-LDS if available, else into
// VGPRs) while chunk k feeds the matrix unit.
// ---------------------------------------------------------------------------
__global__ __launch_bounds__(32)
void gemv_wmma_f32(const float* __restrict__ W, const float* __restrict__ x,
                   const float* __restrict__ bias, float* __restrict__ y,
                   int M, int K, int relu) {
#if GEMV_ASYNC
    __shared__ float  wt[2][16 * 132];
    __shared__ float4 xb4[2][32];
#else
    __shared__ float  wt[1][16 * 132];
    __shared__ float4 xb4[1][32];
#endif

    const int lane = threadIdx.x;     // 0..31
    const int lo = lane & 15;
    const int hi = lane >> 4;
    const int rowBase = blockIdx.x * 16;
    const int nch = K >> 7;           // K / 128

    v8f c = {};

#if GEMV_ASYNC
    // ---- async double-buffer: 17 ASYNCcnt ops per chunk, LDS written by DMA
    auto issue = [&](int p, int k0) {
        __builtin_amdgcn_global_load_async_to_lds_b128(
            to_as1(x + k0 + 4 * lane), to_as3(&xb4[p][lane]), 0, 0);
#pragma unroll
        for (int r = 0; r < 16; ++r) {
            int gr = rowBase + r;
            if (gr >= M) gr = M - 1;                  // clamp (stores guarded)
            __builtin_amdgcn_global_load_async_to_lds_b128(
                to_as1(W + (size_t)gr * K + k0 + 4 * lane),
                to_as3(&wt[p][r * 132 + 4 * lane]), 0, 0);
        }
    };

    issue(0, 0);
    for (int ci = 0; ci < nch; ++ci) {
        const int p = ci & 1;
        if (ci + 1 < nch) {
            // buffer p^1 was read last iteration; its DS reads must land
            // before the async engine rewrites it (async not ordered vs DS)
            asm volatile("s_wait_dscnt 0x0" ::: "memory");
            issue(p ^ 1, (ci + 1) << 7);
            asm volatile("s_wait_asynccnt 0x11" ::: "memory");  // chunk p done
        } else {
            asm volatile("s_wait_asynccnt 0x0" ::: "memory");
        }
        const float* wp = wt[p];
        const float* xp = reinterpret_cast<const float*>(&xb4[p][0]);
#pragma unroll
        for (int s = 0; s < 32; ++s) {
            const int kk = 4 * s + 2 * hi;            // K-pair for this lane half
            v2f a = *reinterpret_cast<const v2f*>(&wp[lo * 132 + kk]);
            v2f b;
            float bx = 0.0f, by = 0.0f;
            if (lo == 0) {                            // column N==0 carries x
                bx = xp[kk];
                by = xp[kk + 1];
            }
            b.x = bx;
            b.y = by;
            c = __builtin_amdgcn_wmma_f32_16x16x4_f32(
                    false, a, false, b, (short)0, c, false, false);
        }
    }
#else
    // ---- register-staged software pipeline (fallback)
    float4 wreg[16];
    float4 xreg;
    auto load = [&](int k0) {
        xreg = reinterpret_cast<const float4*>(x + k0)[lane];
#pragma unroll
        for (int r = 0; r < 16; ++r) {
            int gr = rowBase + r;
            if (gr >= M) gr = M - 1;                  // clamp (stores guarded)
            wreg[r] = reinterpret_cast<const float4*>(W + (size_t)gr * K + k0)[lane];
        }
    };

    load(0);
    for (int ci = 0; ci < nch; ++ci) {
        xb4[0][lane] = xreg;
#pragma unroll
        for (int r = 0; r < 16; ++r)
            *reinterpret_cast<float4*>(&wt[0][r * 132 + 4 * lane]) = wreg[r];
        __syncthreads();
        if (ci + 1 < nch) load((ci + 1) << 7);        // in flight during WMMA
        const float* xp = reinterpret_cast<const float*>(&xb4[0][0]);
#pragma unroll
        for (int s = 0; s < 32; ++s) {
            const int kk = 4 * s + 2 * hi;
            v2f a = *reinterpret_cast<const v2f*>(&wt[0][lo * 132 + kk]);
            v2f b;
            float bx = 0.0f, by = 0.0f;
            if (lo == 0) {
                bx = xp[kk];
                by = xp[kk + 1];
            }
            b.x = bx;
            b.y = by;
            c = __builtin_amdgcn_wmma_f32_16x16x4_f32(
                    false, a, false, b, (short)0, c, false, false);
        }
        __syncthreads();
    }
#endif

    // D column 0: lane 0 -> rows rowBase..+7, lane 16 -> rows rowBase+8..+15
    if (lo == 0) {
        const int rbase = rowBase + 8 * hi;
#pragma unroll
        for (int j = 0; j < 8; ++j) {
            const int rr = rbase + j;
            if (rr < M) {
                float v = c[j] + bias[rr];
                if (relu) v = fmaxf(v, 0.0f);
                y[rr] = v;
            }
        }
    }
}

// ---------------------------------------------------------------------------
// Embedding lookup + build concatenated activation vectors.
// vec_eh = [emb[input], h0]   (for attention GEMV)
// vec_ec = [emb[input], ctx]  (ctx filled later by k_attn)
// ---------------------------------------------------------------------------
__global__ void k_embed(const int* __restrict__ input,
                        const float* __restrict__ hidden,
                        const float* __restrict__ emb,
                        float* __restrict__ vec_eh,
                        float* __restrict__ vec_ec) {
    const int idx = input[0];
    for (int h = threadIdx.x; h < H; h += blockDim.x) {
        const float e = emb[(size_t)idx * H + h];
        vec_eh[h] = e;
        vec_ec[h] = e;
        vec_eh[H + h] = hidden[h];
    }
}

// ---------------------------------------------------------------------------
// Softmax over L attention logits, write attn_weights to d_out tail,
// then context[h] = sum_l w[l] * enc[l,h]  (coalesced across h).
// ---------------------------------------------------------------------------
__global__ void k_attn(const float* __restrict__ logits,
                       const float* __restrict__ enc,
                       float* __restrict__ ctx,
                       float* __restrict__ attn_out) {
    __shared__ float red[256];
    __shared__ float w[128];
    const int t = threadIdx.x;

    const float v = (t < L) ? logits[t] : -INFINITY;
    red[t] = v;
    __syncthreads();
    for (int s = 128; s > 0; s >>= 1) {
        if (t < s) red[t] = fmaxf(red[t], red[t + s]);
        __syncthreads();
    }
    const float mx = red[0];
    __syncthreads();

    const float e = (t < L) ? __expf(v - mx) : 0.0f;
    red[t] = e;
    __syncthreads();
    for (int s = 128; s > 0; s >>= 1) {
        if (t < s) red[t] += red[t + s];
        __syncthreads();
    }
    const float inv = 1.0f / red[0];
    if (t < L) {
        const float ww = e * inv;
        w[t] = ww;
        attn_out[t] = ww;
    }
    __syncthreads();

    for (int h = t; h < H; h += blockDim.x) {
        float acc = 0.0f;
        for (int l = 0; l < L; ++l) acc += w[l] * enc[(size_t)l * H + h];
        ctx[h] = acc;
    }
}

// ---------------------------------------------------------------------------
// GRU gate fusion (PyTorch order r,z,n): h_new -> ws and d_out[V..V+H)
// ---------------------------------------------------------------------------
__global__ void k_gru(const float* __restrict__ gi, const float* __restrict__ gh,
                      const float* __restrict__ h0, float* __restrict__ hnew_ws,
                      float* __restrict__ hnew_out) {
    const int t = blockIdx.x * blockDim.x + threadIdx.x;
    if (t < H) {
        const float r = 1.0f / (1.0f + __expf(-(gi[t] + gh[t])));
        const float z = 1.0f / (1.0f + __expf(-(gi[H + t] + gh[H + t])));
        const float n = tanhf(gi[2 * H + t] + r * gh[2 * H + t]);
        const float hn = (1.0f - z) * n + z * h0[t];
        hnew_ws[t] = hn;
        hnew_out[t] = hn;
    }
}

// ---------------------------------------------------------------------------
// Single-block reduction over V logits: write [max, log(sum exp)] to out2.
// ---------------------------------------------------------------------------
__global__ __launch_bounds__(1024)
void k_lse(const float* __restrict__ logits, float* __restrict__ out2) {
    __shared__ float red[1024];
    const int t = threadIdx.x;

    float mx = -INFINITY;
    for (int i = t; i < V; i += 1024) mx = fmaxf(mx, logits[i]);
    red[t] = mx;
    __syncthreads();
    for (int s = 512; s > 0; s >>= 1) {
        if (t < s) red[t] = fmaxf(red[t], red[t + s]);
        __syncthreads();
    }
    const float gmax = red[0];
    __syncthreads();

    float sum = 0.0f;
    for (int i = t; i < V; i += 1024) sum += __expf(logits[i] - gmax);
    red[t] = sum;
    __syncthreads();
    for (int s = 512; s > 0; s >>= 1) {
        if (t < s) red[t] += red[t + s];
        __syncthreads();
    }
    if (t == 0) {
        out2[0] = gmax;
        out2[1] = __logf(red[0]);
    }
}

__global__ void k_logsoftmax(float* __restrict__ logits,
                             const float* __restrict__ mxlse) {
    const int i = blockIdx.x * blockDim.x + threadIdx.x;
    if (i < V) logits[i] = logits[i] - mxlse[0] - mxlse[1];
}

// ---------------------------------------------------------------------------
extern "C" void kernel_launch(void* const* d_in, const int* in_sizes, int n_in,
                              void* d_out, int out_size, void* d_ws, size_t ws_size,
                              hipStream_t stream) {
    const int*   input  = (const int*)d_in[0];
    const float* hidden = (const float*)d_in[1];
    const float* enc    = (const float*)d_in[2];
    const float* emb    = (const float*)d_in[3];
    const float* attn_W = (const float*)d_in[4];
    const float* attn_b = (const float*)d_in[5];
    const float* comb_W = (const float*)d_in[6];
    const float* comb_b = (const float*)d_in[7];
    const float* W_ih   = (const float*)d_in[8];
    const float* W_hh   = (const float*)d_in[9];
    const float* b_ih   = (const float*)d_in[10];
    const float* b_hh   = (const float*)d_in[11];
    const float* out_W  = (const float*)d_in[12];
    const float* out_b  = (const float*)d_in[13];

    float* out = (float*)d_out;           // [V logsoftmax | H h_new | L attn_w]
    float* ws  = (float*)d_ws;

    float* vec_eh   = ws;                 // 2048: [embedded, h0]
    float* vec_ec   = ws + 2048;          // 2048: [embedded, context]
    float* attn_log = ws + 4096;          // 128 (100 used)
    float* xvec     = ws + 4224;          // 1024: relu(combine)
    float* gi       = ws + 5248;          // 3072
    float* gh       = ws + 8320;          // 3072
    float* hnew     = ws + 11392;         // 1024
    float* mxlse    = ws + 12416;         // 2

    // 1) embedding + concat inputs
    k_embed<<<1, 256, 0, stream>>>(input, hidden, emb, vec_eh, vec_ec);
    // 2) attention logits: [L] = vec_eh @ attn_W.T + attn_b
    gemv_wmma_f32<<<(L + 15) / 16, 32, 0, stream>>>(attn_W, vec_eh, attn_b,
                                                    attn_log, L, 2 * H, 0);
    // 3) softmax + context; attn_weights -> d_out tail
    k_attn<<<1, 256, 0, stream>>>(attn_log, enc, vec_ec + H, out + V + H);
    // 4) combine + relu: [H] = vec_ec @ comb_W.T + comb_b
    gemv_wmma_f32<<<H / 16, 32, 0, stream>>>(comb_W, vec_ec, comb_b,
                                             xvec, H, 2 * H, 1);
    // 5) gi = x @ W_ih.T + b_ih
    gemv_wmma_f32<<<(3 * H) / 16, 32, 0, stream>>>(W_ih, xvec, b_ih,
                                                   gi, 3 * H, H, 0);
    // 6) gh = h0 @ W_hh.T + b_hh
    gemv_wmma_f32<<<(3 * H) / 16, 32, 0, stream>>>(W_hh, hidden, b_hh,
                                                   gh, 3 * H, H, 0);
    // 7) GRU gates -> h_new (ws + d_out)
    k_gru<<<(H + 255) / 256, 256, 0, stream>>>(gi, gh, hidden, hnew, out + V);
    // 8) logits = h_new @ out_W.T + out_b  (the 206 MB stream)
    gemv_wmma_f32<<<(V + 15) / 16, 32, 0, stream>>>(out_W, hnew, out_b,
                                                    out, V, H, 0);
    // 9) max / log-sum-exp over V
    k_lse<<<1, 1024, 0, stream>>>(out, mxlse);
    // 10) log_softmax in place
    k_logsoftmax<<<(V + 255) / 256, 256, 0, stream>>>(out, mxlse);
}